// ElasticDeformation_45148696216101
// MI455X (gfx1250) — compile-verified
//
#include <hip/hip_runtime.h>
#include <math.h>

// ---------- constants (exact values for the cubic B-spline prefilter) ----------
#define HH   1024
#define WW   1024
#define CC   16
#define HWSZ (1024*1024)

#define ZP     (-0.26794919243112270647f)   // z = sqrt(3) - 2
#define GAINF  (6.0f)                       // (1-z)(1-1/z) == 6 exactly
#define KLAST  (0.28867513459481287f)       // z/(z^2-1) == 1/(2*sqrt(3)) exactly
#define UXS    ((float)(2.0/1023.0))        // (P-1)/(H-1)

typedef __attribute__((ext_vector_type(2))) float v2f;
typedef __attribute__((ext_vector_type(8))) float v8f;

// ---------- helpers ----------
__device__ __forceinline__ void cubw(float f, float w[4]) {
    float f2 = f * f, f3 = f2 * f;
    w[0] = (1.f - 3.f*f + 3.f*f2 - f3) * (1.f/6.f);
    w[1] = (4.f - 6.f*f2 + 3.f*f3) * (1.f/6.f);
    w[2] = (1.f + 3.f*f + 3.f*f2 - 3.f*f3) * (1.f/6.f);
    w[3] = f3 * (1.f/6.f);
}

// exact reference prefilter for a length-3 sequence (horizon = 3)
__device__ __forceinline__ void filter3(float x0, float x1, float x2,
                                        float& o0, float& o1, float& o2) {
    float c0 = GAINF * x0, c1 = GAINF * x1, c2 = GAINF * x2;
    float cp0 = c0 + ZP * c1 + ZP * ZP * c2;   // sum z^k * c_k
    float cp1 = c1 + ZP * cp0;
    float cp2 = c2 + ZP * cp1;
    float last = KLAST * (cp2 + ZP * cp1);
    o2 = last;
    o1 = ZP * (o2 - cp1);
    o0 = ZP * (o1 - cp0);
}

// ---------- kernel 1a: control-point coefficients (2x3x3) ----------
__global__ void k_dcoef(const float* __restrict__ disp, float* __restrict__ dco) {
    if (threadIdx.x != 0 || blockIdx.x != 0) return;
    float d[2][3][3];
    for (int c = 0; c < 2; c++)
        for (int j = 0; j < 3; j++)
            for (int i = 0; i < 3; i++)
                d[c][j][i] = disp[c*9 + j*3 + i] * 10.0f;   // SIGMA = 10
    // filter along axis 1 (j) then axis 2 (i)
    for (int c = 0; c < 2; c++)
        for (int i = 0; i < 3; i++)
            filter3(d[c][0][i], d[c][1][i], d[c][2][i],
                    d[c][0][i], d[c][1][i], d[c][2][i]);
    for (int c = 0; c < 2; c++)
        for (int j = 0; j < 3; j++)
            filter3(d[c][j][0], d[c][j][1], d[c][j][2],
                    d[c][j][0], d[c][j][1], d[c][j][2]);
    for (int c = 0; c < 2; c++)
        for (int j = 0; j < 3; j++)
            for (int i = 0; i < 3; i++)
                dco[c*9 + j*3 + i] = d[c][j][i];
}

// ---------- kernel 1b: basis table T[1024][4] and B[c][4][1024] = dcoef * Ax^T ----------
__global__ void k_tables(const float* __restrict__ dco,
                         float* __restrict__ T, float* __restrict__ B) {
    int x = blockIdx.x * 256 + threadIdx.x;
    if (x >= 1024) return;
    float u = (float)x * UXS;
    float fi = floorf(u);
    int   i  = (int)fi;
    float f  = u - fi;
    float w[4]; cubw(f, w);
    float t[3] = {0.f, 0.f, 0.f};
    #pragma unroll
    for (int a = 0; a < 4; a++) {
        int j = i + a - 1;                // in [-1, 4]
        int m = (j < 0) ? -j : j;
        m &= 3;                            // mirror period 2n-2 = 4
        if (m >= 3) m = 4 - m;
        t[m] += w[a];
    }
    T[x*4 + 0] = t[0]; T[x*4 + 1] = t[1]; T[x*4 + 2] = t[2]; T[x*4 + 3] = 0.f;
    for (int c = 0; c < 2; c++)
        for (int j = 0; j < 4; j++) {
            float s = 0.f;
            if (j < 3)
                for (int i2 = 0; i2 < 3; i2++)
                    s += dco[c*9 + j*3 + i2] * t[i2];
            B[(c*4 + j)*1024 + x] = s;
        }
}

// ---------- kernel 2: IIR prefilter along H (thread per column; coalesced) ----------
__global__ void k_prefilter_h(const float* __restrict__ img,
                              float* __restrict__ cp, float* __restrict__ coef) {
    int tid = blockIdx.x * 256 + threadIdx.x;     // 0..16383
    int c = tid >> 10, x = tid & 1023;
    const float* src = img  + (size_t)c * HWSZ + x;
    float*       cpw = cp   + (size_t)c * HWSZ + x;
    float*       dst = coef + (size_t)c * HWSZ + x;
    // c0 = sum_{k<30} z^k * gain * x[k]
    float acc = 0.f, w = GAINF;
    #pragma unroll 5
    for (int k = 0; k < 30; k++) { acc = fmaf(w, src[k*WW], acc); w *= ZP; }
    float prev2 = 0.f, prev = acc;
    cpw[0] = acc;
    for (int y = 1; y < HH; y++) {
        float v = fmaf(ZP, prev, GAINF * src[y*WW]);
        cpw[y*WW] = v;
        prev2 = prev; prev = v;
    }
    float on = KLAST * (prev + ZP * prev2);
    dst[(HH-1)*WW] = on;
    for (int y = HH - 2; y >= 0; y--) {
        on = ZP * (on - cpw[y*WW]);
        dst[y*WW] = on;
    }
}

// ---------- kernel 3: IIR prefilter along W (thread per row; in-place, L2-resident) ----------
__global__ void k_prefilter_w(float* coef, float* __restrict__ cp) {
    int tid = blockIdx.x * 256 + threadIdx.x;
    int c = tid >> 10, y = tid & 1023;
    float* row = coef + (size_t)c * HWSZ + (size_t)y * WW;
    float* cpw = cp   + (size_t)c * HWSZ + (size_t)y * WW;
    float acc = 0.f, w = GAINF;
    #pragma unroll 5
    for (int k = 0; k < 30; k++) { acc = fmaf(w, row[k], acc); w *= ZP; }
    float prev2 = 0.f, prev = acc;
    cpw[0] = acc;
    for (int x = 1; x < WW; x++) {
        float v = fmaf(ZP, prev, GAINF * row[x]);
        cpw[x] = v;
        prev2 = prev; prev = v;
    }
    float on = KLAST * (prev + ZP * prev2);
    row[WW-1] = on;
    for (int x = WW - 2; x >= 0; x--) {
        on = ZP * (on - cpw[x]);
        row[x] = on;
    }
}

// ---------- kernel 4: dense displacement field via V_WMMA_F32_16X16X4_F32 ----------
// F(y,x) = Ay(H x 3) * dcoef(3x3) * Ax^T(3 x W), padded to K=4.
// One wave per 16x16 tile: D(16x16) = A(16x4) x B(4x16). f32 in/out keeps
// sampling-coordinate precision (f16 WMMA would inject ~0.03px coord error).
__global__ void k_field(const float* __restrict__ T, const float* __restrict__ B,
                        float* __restrict__ sy, float* __restrict__ sx) {
    int tile = blockIdx.x;          // 8192 tiles: 2 channels x 64 x 64
    int c  = tile >> 12;
    int t  = tile & 4095;
    int y0 = (t >> 6) * 16;
    int x0 = (t & 63) * 16;
    int lane = threadIdx.x;         // blockDim = 32, EXEC all-ones for WMMA
    int half = lane >> 4;
    int m    = lane & 15;
    const float* Bc = B + c * 4 * 1024;
    // A 16x4 f32 layout: VGPR0 = K=(0|2), VGPR1 = K=(1|3) selected by lane half
    v2f a, b;
    a.x = T[(y0 + m) * 4 + 2*half];
    a.y = T[(y0 + m) * 4 + 2*half + 1];
    // B 4x16: row K striped across lanes within a VGPR, halves hold K and K+2
    b.x = Bc[(2*half)     * 1024 + x0 + m];
    b.y = Bc[(2*half + 1) * 1024 + x0 + m];
    v8f acc = {};
#if __has_builtin(__builtin_amdgcn_wmma_f32_16x16x4_f32)
    acc = __builtin_amdgcn_wmma_f32_16x16x4_f32(false, a, false, b,
                                                (short)0, acc, false, false);
    float* outp = (c == 0) ? sy : sx;
    #pragma unroll
    for (int r = 0; r < 8; r++) {
        int row = y0 + r + 8*half;        // C/D layout: VGPR r -> M = r | r+8
        int col = x0 + m;
        float base = (c == 0) ? (float)row : (float)col;
        outp[row * WW + col] = base + acc[r];
    }
#else
    // scalar fallback (keeps kernel correct if the f32 WMMA builtin is absent)
    float* outp = (c == 0) ? sy : sx;
    #pragma unroll
    for (int r = 0; r < 8; r++) {
        int row = y0 + r + 8*half;
        int col = x0 + m;
        float v = 0.f;
        for (int k = 0; k < 4; k++) v += T[row*4 + k] * Bc[k*1024 + col];
        float base = (c == 0) ? (float)row : (float)col;
        outp[row * WW + col] = base + v;
    }
#endif
}

// ---------- kernel 5: fused sampling (cubic image + nearest mask) ----------
__global__ void k_eval(const float* __restrict__ coef, const float* __restrict__ mask,
                       const float* __restrict__ sy, const float* __restrict__ sx,
                       float* __restrict__ oimg, float* __restrict__ omask) {
    int x = blockIdx.x * 256 + threadIdx.x;
    int y = blockIdx.y;
    int c = blockIdx.z;
    int p = y * WW + x;
    float syv = sy[p], sxv = sx[p];
    bool valid = (syv >= 0.f) && (syv <= 1023.f) && (sxv >= 0.f) && (sxv <= 1023.f);
    size_t cbase = (size_t)c * HWSZ;
    float mo = 0.f, io = 0.f;
    if (valid) {
        // mask: order-0, jnp.round == round-half-to-even -> rintf
        int ry = (int)rintf(syv); ry = min(max(ry, 0), HH - 1);
        int rx = (int)rintf(sxv); rx = min(max(rx, 0), WW - 1);
        mo = mask[cbase + (size_t)ry * WW + rx];
        // image: 4x4 cubic with mirror boundary (period 2046; taps in [-1,1025])
        float fyi = floorf(syv), fxi = floorf(sxv);
        int iy = (int)fyi, ix = (int)fxi;
        float wy[4], wx[4];
        cubw(syv - fyi, wy);
        cubw(sxv - fxi, wx);
        int mys[4], mxs[4];
        #pragma unroll
        for (int a = 0; a < 4; a++) {
            int j = iy + a - 1; if (j < 0) j = -j; if (j >= HH) j = 2*HH - 2 - j;
            mys[a] = j;
            int k = ix + a - 1; if (k < 0) k = -k; if (k >= WW) k = 2*WW - 2 - k;
            mxs[a] = k;
        }
        const float* cc = coef + cbase;
        #pragma unroll
        for (int a = 0; a < 4; a++) {
            const float* rowp = cc + (size_t)mys[a] * WW;
            float r = wx[0]*rowp[mxs[0]] + wx[1]*rowp[mxs[1]]
                    + wx[2]*rowp[mxs[2]] + wx[3]*rowp[mxs[3]];
            io = fmaf(wy[a], r, io);
        }
    }
    oimg[cbase + p]  = io;
    omask[cbase + p] = mo;
}

// ---------- launch ----------
extern "C" void kernel_launch(void* const* d_in, const int* in_sizes, int n_in,
                              void* d_out, int out_size, void* d_ws, size_t ws_size,
                              hipStream_t stream) {
    const float* img  = (const float*)d_in[0];   // (16,1024,1024)
    const float* mask = (const float*)d_in[1];   // (16,1024,1024)
    const float* disp = (const float*)d_in[2];   // (2,3,3)
    float* out_img  = (float*)d_out;                       // first 16M floats
    float* out_mask = (float*)d_out + (size_t)CC * HWSZ;   // next 16M floats

    float* ws   = (float*)d_ws;
    float* cp   = ws;                               // 16*HW  (causal scratch)
    float* coef = ws + (size_t)16 * HWSZ;           // 16*HW  (coef, in-place pass 2)
    float* T    = ws + (size_t)32 * HWSZ;           // 1024*4
    float* B    = T + 4096;                         // 2*4*1024
    float* dco  = B + 8192;                         // 18 floats
    float* syb  = cp;                               // reuse cp after pass 2
    float* sxb  = cp + HWSZ;

    k_dcoef      <<<1, 1, 0, stream>>>(disp, dco);
    k_tables     <<<4, 256, 0, stream>>>(dco, T, B);
    k_prefilter_h<<<64, 256, 0, stream>>>(img, cp, coef);
    k_prefilter_w<<<64, 256, 0, stream>>>(coef, cp);
    k_field      <<<8192, 32, 0, stream>>>(T, B, syb, sxb);
    dim3 g(WW / 256, HH, CC);
    k_eval       <<<g, 256, 0, stream>>>(coef, mask, syb, sxb, out_img, out_mask);
}